// STMambaLoRAInline_12438225289388
// MI455X (gfx1250) — compile-verified
//
#include <hip/hip_runtime.h>

// Problem constants (from reference)
#define Bn 8
#define Tn 16
#define Pn 197
#define Cn 768
#define Rn 64
#define DSn 16
#define NSEQ (Bn * Pn)      // 1576 sequences
#define MROWS (NSEQ * Tn)   // 25216 rows
#define SCALE_F 0.25f       // ALPHA / R

#define KC 64               // K-chunk staged in LDS for k_down
#define NKC (Cn / KC)       // 12 chunks
#define LDSROW 68           // padded row stride (floats): conflict-free ds_load_b64

typedef float v2f __attribute__((ext_vector_type(2)));
typedef float v8f __attribute__((ext_vector_type(8)));
typedef int i4v __attribute__((vector_size(16)));      // int4, matches builtin pointee
typedef __attribute__((address_space(1))) i4v* g_i4p;  // global int4*
typedef __attribute__((address_space(3))) i4v* l_i4p;  // LDS int4*

#if __has_builtin(__builtin_amdgcn_global_load_async_to_lds_b128) && \
    __has_builtin(__builtin_amdgcn_s_wait_asynccnt)
#define ASYNC_LDS 1
#else
#define ASYNC_LDS 0
#endif

// D = A(16x4,f32) * B(4x16,f32) + C(16x16,f32)
static __device__ __forceinline__ v8f wmma4(v2f a, v2f b, v8f c) {
  return __builtin_amdgcn_wmma_f32_16x16x4_f32(
      /*neg_a=*/false, a, /*neg_b=*/false, b,
      /*c_mod=*/(short)0, c, /*reuse_a=*/false, /*reuse_b=*/false);
}

static __device__ __forceinline__ float softplusf(float x) {
  return x > 20.0f ? x : log1pf(expf(x));
}

// Stage one 64x64 f32 tile of W (rows 0..63, cols kc..kc+63) into LDS with
// padded row stride LDSROW. 128 threads, 8 x 16B chunks per thread.
static __device__ __forceinline__ void stage_b(const float* __restrict__ w,
                                               float* dstbase, int kc, int tid) {
#pragma unroll
  for (int q = 0; q < 8; ++q) {
    const int c = q * 128 + tid;           // 0..1023 chunk id
    const int row = c >> 4;                // 0..63
    const int col = (c & 15) << 2;         // float offset, 16B granules
    const float* g = w + (size_t)row * Cn + kc + col;
    float* d = dstbase + row * LDSROW + col;
#if ASYNC_LDS
    __builtin_amdgcn_global_load_async_to_lds_b128(
        (g_i4p)(uintptr_t)g,
        (l_i4p)(uintptr_t)(unsigned)(uintptr_t)d,
        0, 0);
#else
    *(float4*)d = *(const float4*)g;
#endif
  }
}

// ---------------------------------------------------------------------------
// Kernel 1: red_seq[n,t,r] = mask[b,t] * sum_c hs[b,t,p,c] * down_w[r,c]
// One wave per m-tile (= one sequence n, rows t=0..15); 4 n-tiles cover R=64.
// down_w K-chunks are double-buffered in LDS via async global->LDS copies,
// shared by all 4 waves of the block.
// ---------------------------------------------------------------------------
__global__ void __launch_bounds__(128) k_down(const float* __restrict__ hs,
                                              const unsigned char* __restrict__ fmask,
                                              const float* __restrict__ down_w,
                                              float* __restrict__ red_seq) {
  __shared__ float sB[2][64 * LDSROW];   // 2 x 17KB

  const int tid = threadIdx.x;
  const int lane = tid & 31;
  const int wave = tid >> 5;
  const int n = blockIdx.x * 4 + wave;        // sequence / m-tile id (0..1575)
  const int l = lane & 15;
  const int hi = lane >> 4;
  const int b = n / Pn;
  const int p = n % Pn;

  // A row for WMMA lane layout: M = l (= t), K pair at offset 2*hi
  const float* arow = hs + (size_t)(((b * Tn + l) * Pn + p)) * Cn + 2 * hi;

  v8f acc[4] = {};
  stage_b(down_w, &sB[0][0], 0, tid);

  for (int i = 0; i < NKC; ++i) {
    const int kc = i * KC;
    if (i + 1 < NKC) {
      stage_b(down_w, &sB[(i + 1) & 1][0], kc + KC, tid);
#if ASYNC_LDS
      __builtin_amdgcn_s_wait_asynccnt(8);   // chunk i's 8 copies complete
#endif
    } else {
#if ASYNC_LDS
      __builtin_amdgcn_s_wait_asynccnt(0);
#endif
    }
    __syncthreads();                          // all waves' copies visible

    // prefetch the A stream two chunks ahead
    if (kc + 2 * KC < Cn) __builtin_prefetch(arow + kc + 2 * KC, 0, 0);

    const float* bbuf = &sB[i & 1][0];
    for (int k0 = 0; k0 < KC; k0 += 4) {
      v2f a = *(const v2f*)(arow + kc + k0);
#pragma unroll
      for (int j = 0; j < 4; ++j) {
        v2f bb = *(const v2f*)(bbuf + (j * 16 + l) * LDSROW + k0 + 2 * hi);
        acc[j] = wmma4(a, bb, acc[j]);
      }
    }
    __syncthreads();                          // done reading before overwrite
  }

#pragma unroll
  for (int j = 0; j < 4; ++j) {
#pragma unroll
    for (int v = 0; v < 8; ++v) {
      const int t = v + 8 * hi;                      // output row in tile
      const float m = fmask[b * Tn + t] ? 1.0f : 0.0f;
      red_seq[(size_t)(n * Tn + t) * Rn + j * 16 + l] = acc[j][v] * m;
    }
  }
}

// ---------------------------------------------------------------------------
// Kernel 2: proj = red_seq @ x_proj_w^T + b  ->  delta (softplus), Bm, Cm
// ---------------------------------------------------------------------------
__global__ void k_xproj(const float* __restrict__ xseq,
                        const float* __restrict__ w,
                        const float* __restrict__ bias,
                        float* __restrict__ dlt,
                        float* __restrict__ Bm,
                        float* __restrict__ Cm) {
  const int lane = threadIdx.x & 31;
  const int wave = threadIdx.x >> 5;
  const int mt = blockIdx.x * 4 + wave;       // m-tile id
  const int l = lane & 15;
  const int hi = lane >> 4;

  const float* arow = xseq + (size_t)(mt * 16 + l) * Rn + 2 * hi;
  const float* brow = w + (size_t)l * Rn + 2 * hi;

  v8f acc[6] = {};
  for (int k0 = 0; k0 < Rn; k0 += 4) {
    v2f a = *(const v2f*)(arow + k0);
#pragma unroll
    for (int j = 0; j < 6; ++j) {
      v2f bb = *(const v2f*)(brow + (size_t)(j * 16) * Rn + k0);
      acc[j] = wmma4(a, bb, acc[j]);
    }
  }

#pragma unroll
  for (int j = 0; j < 6; ++j) {
    const int o = j * 16 + l;
    const float bo = bias[o];
#pragma unroll
    for (int v = 0; v < 8; ++v) {
      const int m = mt * 16 + v + 8 * hi;
      const float val = acc[j][v] + bo;
      if (j < 4) {
        dlt[(size_t)m * Rn + o] = softplusf(val);
      } else if (j == 4) {
        Bm[(size_t)m * DSn + l] = val;
      } else {
        Cm[(size_t)m * DSn + l] = val;
      }
    }
  }
}

// ---------------------------------------------------------------------------
// Kernel 3: sequential SSM scan. One block (64 threads) per sequence;
// thread r keeps h[r, 0..15] in registers, B/C shared via LDS.
// ---------------------------------------------------------------------------
__global__ void k_scan(const float* __restrict__ xseq,
                       const float* __restrict__ dlt,
                       const float* __restrict__ Bm,
                       const float* __restrict__ Cm,
                       const float* __restrict__ A_log,
                       const float* __restrict__ Dp,
                       float* __restrict__ y) {
  const int n = blockIdx.x;
  const int r = threadIdx.x;   // 0..63
  __shared__ float sB[DSn];
  __shared__ float sC[DSn];

  float A[DSn];
#pragma unroll
  for (int s = 0; s < DSn; ++s) A[s] = -expf(A_log[r * DSn + s]);
  const float Dr = Dp[r];

  float h[DSn];
#pragma unroll
  for (int s = 0; s < DSn; ++s) h[s] = 0.0f;

  for (int t = 0; t < Tn; ++t) {
    const size_t m = (size_t)n * Tn + t;
    if (r < DSn) sB[r] = Bm[m * DSn + r];
    else if (r < 2 * DSn) sC[r - DSn] = Cm[m * DSn + (r - DSn)];
    __syncthreads();

    const float d  = dlt[m * Rn + r];
    const float xv = xseq[m * Rn + r];
    const float dx = d * xv;
    float accv = 0.0f;
#pragma unroll
    for (int s = 0; s < DSn; ++s) {
      h[s] = expf(d * A[s]) * h[s] + dx * sB[s];
      accv += h[s] * sC[s];
    }
    y[m * Rn + r] = accv + Dr * xv;
    __syncthreads();
  }
}

// ---------------------------------------------------------------------------
// Kernel 4: z = y @ out_proj_w^T + out_proj_b   (kept in (n,t,r) layout)
// ---------------------------------------------------------------------------
__global__ void k_outproj(const float* __restrict__ y,
                          const float* __restrict__ w,
                          const float* __restrict__ bias,
                          float* __restrict__ z) {
  const int lane = threadIdx.x & 31;
  const int wave = threadIdx.x >> 5;
  const int mt = blockIdx.x * 4 + wave;
  const int l = lane & 15;
  const int hi = lane >> 4;

  const float* arow = y + (size_t)(mt * 16 + l) * Rn + 2 * hi;
  const float* brow = w + (size_t)l * Rn + 2 * hi;

  v8f acc[4] = {};
  for (int k0 = 0; k0 < Rn; k0 += 4) {
    v2f a = *(const v2f*)(arow + k0);
#pragma unroll
    for (int j = 0; j < 4; ++j) {
      v2f bb = *(const v2f*)(brow + (size_t)(j * 16) * Rn + k0);
      acc[j] = wmma4(a, bb, acc[j]);
    }
  }

#pragma unroll
  for (int j = 0; j < 4; ++j) {
    const int o = j * 16 + l;
    const float bo = bias[o];
#pragma unroll
    for (int v = 0; v < 8; ++v) {
      const int m = mt * 16 + v + 8 * hi;
      z[(size_t)m * Rn + o] = acc[j][v] + bo;
    }
  }
}

// ---------------------------------------------------------------------------
// Kernel 5: out[b,t,p,o] = mask[b,t]*SCALING * sum_r z[n,t,r]*up_w[o,r]
// Each wave: one m-tile (= sequence n) x one 64-wide column group (12 total).
// ---------------------------------------------------------------------------
__global__ void k_up(const float* __restrict__ z,
                     const float* __restrict__ up_w,
                     const unsigned char* __restrict__ fmask,
                     float* __restrict__ out) {
  const int lane = threadIdx.x & 31;
  const int wflat = blockIdx.x * 4 + (threadIdx.x >> 5);
  const int n = wflat / 12;        // sequence / m-tile
  const int ng = wflat % 12;       // 64-wide column group
  const int l = lane & 15;
  const int hi = lane >> 4;

  const float* arow = z + (size_t)(n * 16 + l) * Rn + 2 * hi;
  const float* brow = up_w + (size_t)(ng * 64 + l) * Rn + 2 * hi;

  v8f acc[4] = {};
  for (int k0 = 0; k0 < Rn; k0 += 4) {
    v2f a = *(const v2f*)(arow + k0);
#pragma unroll
    for (int j = 0; j < 4; ++j) {
      v2f bb = *(const v2f*)(brow + (size_t)(j * 16) * Rn + k0);
      acc[j] = wmma4(a, bb, acc[j]);
    }
  }

  const int b = n / Pn;
  const int p = n % Pn;
#pragma unroll
  for (int j = 0; j < 4; ++j) {
    const int col = ng * 64 + j * 16 + l;
#pragma unroll
    for (int v = 0; v < 8; ++v) {
      const int t = v + 8 * hi;
      const float s = fmask[b * Tn + t] ? SCALE_F : 0.0f;
      out[((size_t)(b * Tn + t) * Pn + p) * Cn + col] = acc[j][v] * s;
    }
  }
}

// ---------------------------------------------------------------------------
extern "C" void kernel_launch(void* const* d_in, const int* in_sizes, int n_in,
                              void* d_out, int out_size, void* d_ws, size_t ws_size,
                              hipStream_t stream) {
  const float* hs            = (const float*)d_in[0];
  const unsigned char* fmask = (const unsigned char*)d_in[1];
  // d_in[2] = num_frames (unused; T fixed at 16)
  const float* down_w     = (const float*)d_in[3];
  const float* A_log      = (const float*)d_in[4];
  const float* Dp         = (const float*)d_in[5];
  const float* x_proj_w   = (const float*)d_in[6];
  const float* x_proj_b   = (const float*)d_in[7];
  const float* out_proj_w = (const float*)d_in[8];
  const float* out_proj_b = (const float*)d_in[9];
  const float* up_w       = (const float*)d_in[10];
  float* out = (float*)d_out;

  // Workspace partition (floats): ~29 MB total
  float* ws = (float*)d_ws;
  size_t off = 0;
  float* red_seq = ws + off; off += (size_t)MROWS * Rn;   // (n,t,r)
  float* dlt     = ws + off; off += (size_t)MROWS * Rn;   // softplus(delta)
  float* Bmb     = ws + off; off += (size_t)MROWS * DSn;
  float* Cmb     = ws + off; off += (size_t)MROWS * DSn;
  float* yb      = ws + off; off += (size_t)MROWS * Rn;   // scan output
  float* zb      = ws + off;                              // out-proj output

  const int mtiles = NSEQ;              // 1576 m-tiles of 16 rows
  const int blk = 128;                  // 4 waves per block

  k_down<<<mtiles / 4, blk, 0, stream>>>(hs, fmask, down_w, red_seq);
  k_xproj<<<mtiles / 4, blk, 0, stream>>>(red_seq, x_proj_w, x_proj_b, dlt, Bmb, Cmb);
  k_scan<<<NSEQ, 64, 0, stream>>>(red_seq, dlt, Bmb, Cmb, A_log, Dp, yb);
  k_outproj<<<mtiles / 4, blk, 0, stream>>>(yb, out_proj_w, out_proj_b, zb);
  k_up<<<mtiles * 12 / 4, blk, 0, stream>>>(zb, up_w, fmask, out);
}